// FineMatching_58669253263497
// MI455X (gfx1250) — compile-verified
//
#include <hip/hip_runtime.h>

// CDNA5 (gfx1250) fine-matching kernel.
// Per-m 64x64x64 f32 GEMM via V_WMMA_F32_16X16X4_F32 (wave32),
// staged through LDS with async global->LDS copies (ASYNCcnt path);
// double-softmax heatmap; argmax; reg refinement.

typedef __attribute__((ext_vector_type(2))) float v2f;
typedef __attribute__((ext_vector_type(8))) float v8f;

#define NCH     128     // C
#define CLS_C2  64      // CLS_C
#define W1      10      // CLS_W + 2*PAD
#define PITCH   68      // LDS row pitch in floats (bank-conflict-free WMMA loads)

// ---- async global->LDS b128 copy (gfx1250 GLOBAL_LOAD_ASYNC_TO_LDS_B128) ----
// Probe-learned signature: (V4i addrspace(1)*, V4i addrspace(3)*, imm int, imm int)
#if __has_builtin(__builtin_amdgcn_global_load_async_to_lds_b128)
#define FM_ASYNC_LDS 1
typedef int fm_v4i __attribute__((vector_size(16)));   // matches builtin's V4i param
__device__ static inline void fm_copy_b128(const void* gsrc, void* lds_dst) {
    // AS1 ptr: generic global pointer value == global address.
    // AS3 ptr: low 32 bits of generic shared pointer == LDS byte offset.
    __builtin_amdgcn_global_load_async_to_lds_b128(
        (__attribute__((address_space(1))) fm_v4i*)(uintptr_t)gsrc,
        (__attribute__((address_space(3))) fm_v4i*)(uint32_t)(uintptr_t)lds_dst,
        /*offset=*/0, /*cpol=*/0);
}
__device__ static inline void fm_wait_async() {
#if __has_builtin(__builtin_amdgcn_s_wait_asynccnt)
    __builtin_amdgcn_s_wait_asynccnt(0);
#else
    asm volatile("s_wait_asynccnt 0" ::: "memory");
#endif
}
#else
#define FM_ASYNC_LDS 0
__device__ static inline void fm_copy_b128(const void* gsrc, void* lds_dst) {
    *(float4*)lds_dst = *(const float4*)gsrc;   // global->VGPR->LDS fallback
}
__device__ static inline void fm_wait_async() {}
#endif

__global__ __launch_bounds__(128) void fine_matching_kernel(
    const float* __restrict__ f0, const float* __restrict__ f1,
    float* __restrict__ out, int M)
{
    __shared__ float sA[64 * PITCH];   // cls_f0  (64 pos x 64 ch)
    __shared__ float sB[64 * PITCH];   // cls_f1m (64 pos x 64 ch)
    __shared__ float sS[64 * PITCH];   // sim     (64 x 64)
    __shared__ float sRmax[64], sRsum[64];   // softmax over axis=2 (per l)
    __shared__ float sCmax[64], sCsum[64];   // softmax over axis=1 (per s)
    __shared__ float sBestV[128];
    __shared__ int   sBestI[128];
    __shared__ int   sBest;
    __shared__ float sSimR[9];

    const int m = blockIdx.x;
    const int t = threadIdx.x;

    const float* f0m = f0 + (size_t)m * 64  * NCH;
    const float* f1m = f1 + (size_t)m * 100 * NCH;

    // ---------- stage cls halves into LDS (async b128, coalesced) ----------
    #pragma unroll
    for (int q = t; q < 64 * 16; q += 128) {          // 8 iters/thread
        int row = q >> 4, c4 = q & 15;
        fm_copy_b128(f0m + row * NCH + c4 * 4,
                     &sA[row * PITCH + c4 * 4]);      // pitch*4 = 272B, 16B aligned
    }
    #pragma unroll
    for (int q = t; q < 64 * 16; q += 128) {
        int row = q >> 4, c4 = q & 15;
        int pos = ((row >> 3) + 1) * W1 + ((row & 7) + 1);   // inner 8x8 of 10x10
        fm_copy_b128(f1m + pos * NCH + c4 * 4,
                     &sB[row * PITCH + c4 * 4]);
    }
    fm_wait_async();           // own wave's async transfers landed in LDS
    __syncthreads();           // all waves' transfers visible

    // ---------- 64x64x64 GEMM: sim = A * B^T / 64 ----------
    // A 16x4 f32 layout: lane%16 = M row; lane/16 selects K pair {0,1}/{2,3}.
    // B 4x16 mirrors A along N (B[k][n] = cls_f1m[n][k] since sim = f0 @ f1m^T).
    const int lane  = t & 31;
    const int wv    = t >> 5;        // 4 waves: wave wv owns rows 16*wv..16*wv+15
    const int mrow  = lane & 15;
    const int khalf = lane >> 4;

    v2f a_k[16];
    {
        const float* ab = &sA[(wv * 16 + mrow) * PITCH + 2 * khalf];
        #pragma unroll
        for (int k = 0; k < 16; ++k) a_k[k] = *(const v2f*)(ab + 4 * k);
    }
    #pragma unroll
    for (int tc = 0; tc < 4; ++tc) {
        const float* bb = &sB[(tc * 16 + mrow) * PITCH + 2 * khalf];
        v8f acc = {0.f, 0.f, 0.f, 0.f, 0.f, 0.f, 0.f, 0.f};
        #pragma unroll
        for (int k = 0; k < 16; ++k) {
            v2f b = *(const v2f*)(bb + 4 * k);
            acc = __builtin_amdgcn_wmma_f32_16x16x4_f32(
                /*neg_a=*/false, a_k[k], /*neg_b=*/false, b,
                /*c_mod=*/(short)0, acc, /*reuse_a=*/false, /*reuse_b=*/false);
        }
        // C/D layout: VGPR r -> row r + 8*(lane>=16), col = lane%16
        #pragma unroll
        for (int r = 0; r < 8; ++r) {
            int row = wv * 16 + khalf * 8 + r;
            int col = tc * 16 + mrow;
            sS[row * PITCH + col] = acc[r] * (1.0f / 64.0f);   // / CLS_C / CLS_T
        }
    }
    __syncthreads();

    // ---------- softmax statistics ----------
    if (t < 64) {                      // per-row (axis=2) stats for l = t
        const float* rp = &sS[t * PITCH];
        float mx = rp[0];
        for (int c = 1; c < 64; ++c) mx = fmaxf(mx, rp[c]);
        float sm = 0.f;
        for (int c = 0; c < 64; ++c) sm += expf(rp[c] - mx);
        sRmax[t] = mx; sRsum[t] = sm;
    } else {                           // per-column (axis=1) stats for s = t-64
        int c = t - 64;
        float mx = sS[c];
        for (int l = 1; l < 64; ++l) mx = fmaxf(mx, sS[l * PITCH + c]);
        float sm = 0.f;
        for (int l = 0; l < 64; ++l) sm += expf(sS[l * PITCH + c] - mx);
        sCmax[c] = mx; sCsum[c] = sm;
    }
    __syncthreads();

    // ---------- heatmap write (coalesced) + argmax ----------
    float* heat = out + (size_t)m * 4096;
    float bv = -1.f;                   // heatmap >= 0
    int   bi = 0;
    #pragma unroll
    for (int i = 0; i < 32; ++i) {
        int e = i * 128 + t;
        int l = e >> 6, c = e & 63;
        float s = sS[l * PITCH + c];
        float h = (expf(s - sCmax[c]) / sCsum[c]) * (expf(s - sRmax[l]) / sRsum[l]);
        heat[e] = h;
        if (h > bv) { bv = h; bi = e; }
    }
    sBestV[t] = bv; sBestI[t] = bi;
    __syncthreads();
    if (t == 0) {
        float v = sBestV[0]; int idx = sBestI[0];
        for (int j = 1; j < 128; ++j) {
            float vj = sBestV[j]; int ij = sBestI[j];
            if (vj > v || (vj == v && ij < idx)) { v = vj; idx = ij; }
        }
        sBest = idx;
    }
    __syncthreads();

    // ---------- reg refinement ----------
    const int best = sBest;
    const int idx0 = best >> 6;
    const int idx1 = best & 63;
    if (t < 9) {
        int i1 = idx1 >> 3, j1 = idx1 & 7;
        int pr = i1 + t / 3, pc = j1 + t % 3;         // 3x3 patch in 10x10 grid
        const float* a = f0m + idx0 * NCH + CLS_C2;
        const float* b = f1m + (pr * W1 + pc) * NCH + CLS_C2;
        float d = 0.f;
        #pragma unroll
        for (int c = 0; c < 64; c += 4) {
            float4 av = *(const float4*)(a + c);
            float4 bw = *(const float4*)(b + c);
            d += av.x * bw.x + av.y * bw.y + av.z * bw.z + av.w * bw.w;
        }
        sSimR[t] = d * (1.0f / 8.0f);                 // / sqrt(64) / REG_T
    }
    __syncthreads();
    if (t == 0) {
        float mx = sSimR[0];
        for (int r = 1; r < 9; ++r) mx = fmaxf(mx, sSimR[r]);
        float e9[9]; float sm = 0.f;
        for (int r = 0; r < 9; ++r) { e9[r] = expf(sSimR[r] - mx); sm += e9[r]; }
        float bx = 0.f, by = 0.f;
        for (int r = 0; r < 9; ++r) {
            float h = e9[r] / sm;
            bx += h * (float)((r % 3) - 1);           // gx = lin[col]
            by += h * (float)((r / 3) - 1);           // gy = lin[row]
        }
        // outputs flat in tuple order (all as float)
        size_t base = (size_t)M * 4096;
        out[base + m]      = (float)idx0;                     // idxes0
        out[base + M + m]  = (float)idx1;                     // idxes1
        float* b0 = out + base + (size_t)2 * M + (size_t)2 * m;
        b0[0] = (float)(idx0 & 7) - 3.5f;                     // biases0 = (x-3.5, y-3.5)
        b0[1] = (float)(idx0 >> 3) - 3.5f;
        float* b1 = out + base + (size_t)4 * M + (size_t)2 * m;
        b1[0] = (float)(idx1 & 7) - 3.5f;
        b1[1] = (float)(idx1 >> 3) - 3.5f;
        float* rb = out + base + (size_t)6 * M + (size_t)2 * m;
        rb[0] = bx; rb[1] = by;
    }
}

extern "C" void kernel_launch(void* const* d_in, const int* in_sizes, int n_in,
                              void* d_out, int out_size, void* d_ws, size_t ws_size,
                              hipStream_t stream) {
    const float* f0 = (const float*)d_in[0];   // (M, 64, 128) f32
    const float* f1 = (const float*)d_in[1];   // (M, 100, 128) f32
    float* out = (float*)d_out;
    int M = in_sizes[0] / (64 * 128);
    fine_matching_kernel<<<M, 128, 0, stream>>>(f0, f1, out, M);
}